// Skipgram_25984552140867
// MI455X (gfx1250) — compile-verified
//
#include <hip/hip_runtime.h>
#include <hip/hip_bf16.h>
#include <stdint.h>

// Problem constants (match reference)
#define VOCAB 100000
#define DIM   128
#define BATCH 16384
#define KNEG  20

// Launch shape: 2048 blocks x 64 threads = 4096 waves, 4 samples per wave,
// double-buffered LDS staging per wave.
#define THREADS 64
#define WAVES_PER_BLOCK 2
#define NBLK 2048
#define TOTAL_WAVES (NBLK * WAVES_PER_BLOCK)   // 4096
#define ITERS (BATCH / TOTAL_WAVES)            // 4 (exact)

// Per-buffer LDS layout (one buffer = one sample's working set):
//   rows 0..20  : gathered out_embed rows (row 0 = context, 1..20 = negatives)
//   row  21     : zero pad row (tile-1 B-operand for vectors 21..31)
//   offset T_OFF: the 512B target row t
// Row stride 528 B = 132 dwords == 4 (mod 64 banks) -> conflict-free ds_load_b64.
#define ROW_STRIDE 528
#define ZERO_ROW   21
#define T_OFF      (22 * ROW_STRIDE)
#define BUF_STRIDE 12160                 // 22*528 + 512 = 12128, padded; 16B aligned
#define WAVE_LDS   (2 * BUF_STRIDE)      // double buffer

typedef __attribute__((ext_vector_type(2))) float v2f;
typedef __attribute__((ext_vector_type(8))) float v8f;

// Fast, branch-free, numerically-stable log-sigmoid using the hardware
// v_exp_f32 / v_log_f32. exp(-|x|) is in (0,1]; absolute error of
// log(1+e) vs log1p(e) is < ~1e-7, negligible for a mean loss.
__device__ __forceinline__ float log_sigmoid_fast(float x) {
    return fminf(x, 0.0f) - __logf(1.0f + __expf(-fabsf(x)));
}

// Issue the 22 async row gathers (ctx, 20 negs, t) for sample b into the
// LDS buffer at byte offset lbuf. All 22 indices are read into registers
// FIRST (the asm "memory" clobbers would otherwise serialize each s_load
// against the previous async issue); the 20 consecutive neg indices batch
// into wide scalar loads with a single kmcnt wait. Then one async
// instruction per row moves 512B (32 lanes x 16B), fully coalesced.
__device__ __forceinline__ void stage_sample(uint32_t lbuf, int lane, int b,
                                             const int* __restrict__ target,
                                             const int* __restrict__ context,
                                             const int* __restrict__ negs,
                                             const float* __restrict__ in_embed,
                                             const float* __restrict__ out_embed)
{
    int idx[21];
    idx[0] = context[b];                 // row 0
    #pragma unroll
    for (int j = 0; j < KNEG; ++j)       // rows 1..20 (consecutive dwords)
        idx[1 + j] = negs[b * KNEG + j];
    const int tidx = target[b];          // t row

    const uint64_t laneByte = (uint64_t)(lane * 16);
    #pragma unroll
    for (int j = 0; j < 21; ++j) {
        uint64_t g = (uint64_t)(uintptr_t)(out_embed + (size_t)idx[j] * DIM) + laneByte;
        uint32_t l = lbuf + j * ROW_STRIDE + lane * 16;
        asm volatile("global_load_async_to_lds_b128 %0, %1, off" :: "v"(l), "v"(g) : "memory");
    }
    {
        uint64_t g = (uint64_t)(uintptr_t)(in_embed + (size_t)tidx * DIM) + laneByte;
        uint32_t l = lbuf + T_OFF + lane * 16;
        asm volatile("global_load_async_to_lds_b128 %0, %1, off" :: "v"(l), "v"(g) : "memory");
    }
}

__global__ __launch_bounds__(THREADS)
void skipgram_loss_kernel(const int* __restrict__ target,
                          const int* __restrict__ context,
                          const int* __restrict__ negs,
                          const float* __restrict__ in_embed,
                          const float* __restrict__ out_embed,
                          float* __restrict__ partials)
{
    __shared__ __align__(1024) char smem[WAVES_PER_BLOCK * WAVE_LDS];
    __shared__ float wsum[WAVES_PER_BLOCK];

    const int lane = threadIdx.x & 31;
    const int wave = threadIdx.x >> 5;
    char* wbase = smem + wave * WAVE_LDS;
    // Low 32 bits of a generic pointer into the LDS aperture == LDS byte
    // offset, which is what the async-load VDST VGPR wants.
    const uint32_t lbase = (uint32_t)(uintptr_t)wbase;

    // Zero both buffers' pad rows once; async stages never overwrite them.
    #pragma unroll
    for (int pb = 0; pb < 2; ++pb)
        for (int d = lane; d < ROW_STRIDE / 4; d += 32)
            ((uint32_t*)(wbase + pb * BUF_STRIDE + ZERO_ROW * ROW_STRIDE))[d] = 0u;

    const int hi = lane >> 4;       // which K-half of the 16x4 WMMA chunk
    const int m  = lane & 15;       // row/column index within a 16-tile
    const int r2 = (m <= 4) ? (16 + m) : ZERO_ROW;  // tile-1 rows: vectors 16..20, else zeros

    // Wave-uniform sample index (readfirstlane lets index loads be s_loads).
    const int b0 = __builtin_amdgcn_readfirstlane((int)(blockIdx.x * WAVES_PER_BLOCK + wave));

    float lossAcc = 0.0f;

    // Prologue: stage the first sample into buffer 0.
    stage_sample(lbase, lane, b0, target, context, negs, in_embed, out_embed);

    #pragma unroll
    for (int i = 0; i < ITERS; ++i) {
        const char* buf = wbase + (i & 1) * BUF_STRIDE;

        // Stage the NEXT sample into the other buffer before computing this
        // one. WAR: ds reads that targeted that buffer (iteration i-1) must
        // retire first.
        if (i + 1 < ITERS) {
            asm volatile("s_wait_dscnt 0x0" ::: "memory");
            stage_sample(lbase + ((i + 1) & 1) * BUF_STRIDE, lane,
                         b0 + (i + 1) * TOTAL_WAVES,
                         target, context, negs, in_embed, out_embed);
            // 22 newer loads may be outstanding; async loads complete in
            // order, so ASYNCcnt <= 22 means the current buffer is ready.
            asm volatile("s_wait_asynccnt 0x16" ::: "memory");
        } else {
            asm volatile("s_wait_asynccnt 0x0" ::: "memory");
        }

        // A operand: t replicated into all 16 rows (per-half broadcast read);
        // B operands: lane n reads vector n's K-chunk. Every row of D then
        // equals the 16 scores, so acc[0] holds score[lane&15] in every lane.
        const char* aBase  = buf + T_OFF + hi * 8;
        const char* b0Base = buf + m  * ROW_STRIDE + hi * 8;
        const char* b1Base = buf + r2 * ROW_STRIDE + hi * 8;

        v8f acc0 = {};   // vectors 0..15
        v8f acc1 = {};   // vectors 16..20 (+zero pad)
        #pragma unroll 4
        for (int s = 0; s < DIM / 4; ++s) {
            v2f a   = *(const v2f*)(aBase  + 16 * s);
            v2f bb0 = *(const v2f*)(b0Base + 16 * s);
            v2f bb1 = *(const v2f*)(b1Base + 16 * s);
            acc0 = __builtin_amdgcn_wmma_f32_16x16x4_f32(false, a, false, bb0,
                                                         (short)0, acc0, false, false);
            acc1 = __builtin_amdgcn_wmma_f32_16x16x4_f32(false, a, false, bb1,
                                                         (short)0, acc1, false, false);
        }

        float s0 = acc0[0];   // score of vector m      (replicated in both halves)
        float s1 = acc1[0];   // score of vector 16 + m (valid only for m <= 4)
        // Branch-free: flip the argument sign for negatives, mask pad lanes.
        float arg0 = (m == 0) ? s0 : -s0;            // v_cndmask on input
        float msk1 = (m <= 4) ? 1.0f : 0.0f;
        lossAcc += log_sigmoid_fast(arg0) + msk1 * log_sigmoid_fast(-s1);
    }

    // Wave reduction (wave32) then block partial.
    #pragma unroll
    for (int o = 16; o > 0; o >>= 1)
        lossAcc += __shfl_xor(lossAcc, o, 32);
    if (lane == 0) wsum[wave] = lossAcc;
    __syncthreads();
    if (threadIdx.x == 0)
        partials[blockIdx.x] = wsum[0] + wsum[1];
}

// Deterministic fixed-order final reduction (no float atomics -> bit-stable
// across graph replays).
__global__ __launch_bounds__(256)
void skipgram_reduce_kernel(const float* __restrict__ partials, float* __restrict__ out)
{
    __shared__ float sh[256];
    float s = 0.0f;
    for (int i = threadIdx.x; i < NBLK; i += 256)
        s += partials[i];
    sh[threadIdx.x] = s;
    __syncthreads();
    #pragma unroll
    for (int o = 128; o > 0; o >>= 1) {
        if ((int)threadIdx.x < o) sh[threadIdx.x] += sh[threadIdx.x + o];
        __syncthreads();
    }
    if (threadIdx.x == 0)
        out[0] = -sh[0] / (2.0f * (float)BATCH);   // /2 undoes lane-half duplication
}

extern "C" void kernel_launch(void* const* d_in, const int* in_sizes, int n_in,
                              void* d_out, int out_size, void* d_ws, size_t ws_size,
                              hipStream_t stream) {
    const int*   target  = (const int*)d_in[0];
    const int*   context = (const int*)d_in[1];
    const int*   negs    = (const int*)d_in[2];
    const float* in_e    = (const float*)d_in[3];
    const float* out_e   = (const float*)d_in[4];
    float* partials = (float*)d_ws;          // NBLK floats of scratch
    float* out      = (float*)d_out;

    skipgram_loss_kernel<<<NBLK, THREADS, 0, stream>>>(target, context, negs,
                                                       in_e, out_e, partials);
    skipgram_reduce_kernel<<<1, 256, 0, stream>>>(partials, out);
}